// ClustGeoNodeEncoder_16441134809143
// MI455X (gfx1250) — compile-verified
//
#include <hip/hip_runtime.h>
#include <math.h>

typedef __attribute__((ext_vector_type(2))) float v2f;
typedef __attribute__((ext_vector_type(8))) float v8f;

#define N_CLUSTS 16384
#define P_MAX 128

__device__ __forceinline__ float wsum32(float v) {
#pragma unroll
    for (int o = 16; o > 0; o >>= 1) v += __shfl_xor(v, o, 32);
    return v;
}

__global__ __launch_bounds__(128)
void clust_geo_kernel(const float* __restrict__ data,    // (2e6, 10)
                      const int*   __restrict__ clusts,  // (C, 128)
                      const int*   __restrict__ lengths, // (C,)
                      float*       __restrict__ out)     // (C, 22)
{
    __shared__ __align__(16) float xsh [4][P_MAX][4];  // raw gathered coords
    __shared__ __align__(16) float xcsh[4][P_MAX][4];  // centered*masked coords
    const int lane = threadIdx.x & 31;
    const int wid  = threadIdx.x >> 5;
    const int c    = (int)blockIdx.x * 4 + wid;

    const int   len = lengths[c];
    const float n   = (float)len;

    // ---- gather: point t = j*32 + lane (coalesced index loads) ----
    float rx[4], ry[4], rz[4], mj[4];
#pragma unroll
    for (int j = 0; j < 4; ++j) {
        const int t   = j * 32 + lane;
        const int idx = clusts[c * P_MAX + t];
        const float* row = data + (size_t)idx * 10u;
        rx[j] = row[0]; ry[j] = row[1]; rz[j] = row[2];
        mj[j] = (t < len) ? 1.0f : 0.0f;
        float4 v; v.x = rx[j]; v.y = ry[j]; v.z = rz[j]; v.w = 0.0f;
        *(float4*)&xsh[wid][t][0] = v;
    }

    // ---- masked mean (center) ----
    float sx = 0.f, sy = 0.f, sz = 0.f;
#pragma unroll
    for (int j = 0; j < 4; ++j) { sx += mj[j]*rx[j]; sy += mj[j]*ry[j]; sz += mj[j]*rz[j]; }
    const float inv_n = 1.0f / n;
    const float cx = wsum32(sx) * inv_n;
    const float cy = wsum32(sy) * inv_n;
    const float cz = wsum32(sz) * inv_n;

    // ---- stage xc = (x - center)*mask into LDS (float4, 4th = 0) ----
#pragma unroll
    for (int j = 0; j < 4; ++j) {
        const int t = j * 32 + lane;
        float4 v;
        v.x = (rx[j]-cx)*mj[j]; v.y = (ry[j]-cy)*mj[j];
        v.z = (rz[j]-cz)*mj[j]; v.w = 0.0f;
        *(float4*)&xcsh[wid][t][0] = v;
    }
    __syncthreads();

    // ---- covariance A = xc^T xc via V_WMMA_F32_16X16X4_F32 ----
    // A 16x4 f32 layout: vgpr0 = (M=lane%16, K = 0|2), vgpr1 = (M, K = 1|3).
    // B mirrors with N in place of M; B = A^T here so one register image
    // serves both operands. Lanes with comp>=3 read element 3 (== 0.0f):
    // completely branchless staging, loop-invariant LDS address.
    const int half  = lane >> 4;
    const int comp  = lane & 15;
    const int compc = (comp < 3) ? comp : 3;
    v8f acc = {0.f,0.f,0.f,0.f,0.f,0.f,0.f,0.f};
#pragma unroll
    for (int cc = 0; cc < 32; ++cc) {
        const int k0 = cc * 4 + 2 * half;
        const float a0 = xcsh[wid][k0    ][compc];
        const float a1 = xcsh[wid][k0 + 1][compc];
        v2f av = {a0, a1};
        acc = __builtin_amdgcn_wmma_f32_16x16x4_f32(
            false, av, false, av, (short)0, acc, false, false);
    }
    // D layout: VGPR i holds M=i for lanes 0-15 (N=lane)
    const float a00 = __shfl(acc[0], 0, 32);
    const float a01 = __shfl(acc[0], 1, 32);
    const float a02 = __shfl(acc[0], 2, 32);
    const float a11 = __shfl(acc[1], 1, 32);
    const float a12 = __shfl(acc[1], 2, 32);
    const float a22 = __shfl(acc[2], 2, 32);

    // ---- closed-form symmetric 3x3 eigenvalues (ascending w0<=w1<=w2) ----
    const float q   = (a00 + a11 + a22) * (1.0f/3.0f);
    const float b00 = a00 - q, b11 = a11 - q, b22 = a22 - q;
    const float p2  = b00*b00 + b11*b11 + b22*b22 + 2.0f*(a01*a01 + a02*a02 + a12*a12);
    const float p   = sqrtf(fmaxf(p2, 0.0f) * (1.0f/6.0f));
    float w0, w1, w2;
    if (p > 1e-30f) {
        const float ip  = 1.0f / p;
        const float c00 = b00*ip, c01 = a01*ip, c02 = a02*ip;
        const float c11 = b11*ip, c12 = a12*ip, c22 = b22*ip;
        float r = 0.5f * (c00*(c11*c22 - c12*c12)
                        - c01*(c01*c22 - c12*c02)
                        + c02*(c01*c12 - c11*c02));
        r = fminf(fmaxf(r, -1.0f), 1.0f);
        const float phi = acosf(r) * (1.0f/3.0f);
        w2 = q + 2.0f*p*cosf(phi);
        w0 = q + 2.0f*p*cosf(phi + 2.0943951023931953f);
        w1 = 3.0f*q - w0 - w2;
    } else {
        w0 = w1 = w2 = q;
    }
    (void)w0;

    const float dirwt = (w2 == 0.0f) ? 0.0f : (1.0f - w1 / w2);
    const float iw2   = 1.0f / w2;   // DELTA=0 => B = V diag(w/w2) V^T = A / w2

    // ---- top eigenvector: largest cross product of rows of (A - w2 I) ----
    const float m00 = a00 - w2, m11 = a11 - w2, m22 = a22 - w2;
    const float e0x = a01*a12 - a02*m11, e0y = a02*a01 - m00*a12, e0z = m00*m11 - a01*a01;
    const float e1x = a01*m22 - a02*a12, e1y = a02*a02 - m00*m22, e1z = m00*a12 - a01*a02;
    const float e2x = m11*m22 - a12*a12, e2y = a12*a02 - a01*m22, e2z = a01*a12 - m11*a02;
    const float q0 = e0x*e0x + e0y*e0y + e0z*e0z;
    const float q1 = e1x*e1x + e1y*e1y + e1z*e1z;
    const float q2 = e2x*e2x + e2y*e2y + e2z*e2z;
    float ex, ey, ez, nn;
    if (q0 >= q1 && q0 >= q2)      { ex = e0x; ey = e0y; ez = e0z; nn = q0; }
    else if (q1 >= q2)             { ex = e1x; ey = e1y; ez = e1z; nn = q1; }
    else                           { ex = e2x; ey = e2y; ez = e2z; nn = q2; }
    if (nn > 1e-30f) {
        const float inv = 1.0f / sqrtf(nn);
        ex *= inv; ey *= inv; ez *= inv;
    } else { ex = 0.f; ey = 0.f; ez = 1.f; }

    // ---- sign fix: sc = sum(x0 * np0) ----
    float scp = 0.f;
#pragma unroll
    for (int j = 0; j < 4; ++j) {
        const float xcx = (rx[j]-cx)*mj[j], xcy = (ry[j]-cy)*mj[j], xcz = (rz[j]-cz)*mj[j];
        const float x0  = xcx*ex + xcy*ey + xcz*ez;
        const float px = xcx - x0*ex, py = xcy - x0*ey, pz = xcz - x0*ez;
        const float sq = px*px + py*py + pz*pz;
        const float np0 = sqrtf(mj[j] > 0.f ? sq : 1.0f) * mj[j];
        scp += x0 * np0;
    }
    const float sc  = wsum32(scp);
    const float sgn = (sc < 0.f) ? -1.0f : 1.0f;
    const float v0x = dirwt*sgn*ex, v0y = dirwt*sgn*ey, v0z = dirwt*sgn*ez;
    const float nv  = sqrtf(v0x*v0x + v0y*v0y + v0z*v0z);
    const float nvs = (nv == 0.f) ? 1.0f : nv;

    // ---- argmax / argmin of projection p = x . v0 (first-index ties) ----
    float bvMax, bvMin; int biMax, biMin;
    {
        const float p0 = rx[0]*v0x + ry[0]*v0y + rz[0]*v0z;
        bvMax = (mj[0] > 0.f) ? p0 : -INFINITY;  biMax = lane;
        bvMin = (mj[0] > 0.f) ? p0 :  INFINITY;  biMin = lane;
#pragma unroll
        for (int j = 1; j < 4; ++j) {
            const int   t  = j * 32 + lane;
            const float pv = rx[j]*v0x + ry[j]*v0y + rz[j]*v0z;
            const float pmx = (mj[j] > 0.f) ? pv : -INFINITY;
            const float pmn = (mj[j] > 0.f) ? pv :  INFINITY;
            if (pmx > bvMax) { bvMax = pmx; biMax = t; }
            if (pmn < bvMin) { bvMin = pmn; biMin = t; }
        }
#pragma unroll
        for (int o = 16; o > 0; o >>= 1) {
            float ov = __shfl_xor(bvMax, o, 32); int oi = __shfl_xor(biMax, o, 32);
            if (ov > bvMax || (ov == bvMax && oi < biMax)) { bvMax = ov; biMax = oi; }
            ov = __shfl_xor(bvMin, o, 32); oi = __shfl_xor(biMin, o, 32);
            if (ov < bvMin || (ov == bvMin && oi < biMin)) { bvMin = ov; biMin = oi; }
        }
    }
    const float sp1x = xsh[wid][biMax][0], sp1y = xsh[wid][biMax][1], sp1z = xsh[wid][biMax][2];
    const float sp2x = xsh[wid][biMin][0], sp2y = xsh[wid][biMin][1], sp2z = xsh[wid][biMin][2];

    // ---- angle sums toward each candidate endpoint ----
    float m1p = 0.f, m2p = 0.f;
#pragma unroll
    for (int j = 0; j < 4; ++j) {
        const float d1x = rx[j]-sp1x, d1y = ry[j]-sp1y, d1z = rz[j]-sp1z;
        const float d2x = rx[j]-sp2x, d2y = ry[j]-sp2y, d2z = rz[j]-sp2z;
        const float s1 = d1x + d1y + d1z;
        const float s2 = d2x + d2y + d2z;
        const bool valid = (mj[j] > 0.f) && (fabsf(s1) != 0.f) && (fabsf(s2) != 0.f);
        if (valid) {
            const float nd1 = sqrtf(d1x*d1x + d1y*d1y + d1z*d1z);
            float c1 = (d1x*v0x + d1y*v0y + d1z*v0z) / (nd1 * nvs);
            c1 = fminf(fmaxf(c1, -1.f), 1.f);
            m1p += acosf(fabsf(c1));
            const float nd2 = sqrtf(d2x*d2x + d2y*d2y + d2z*d2z);
            float c2 = (d2x*v0x + d2y*v0y + d2z*v0z) / (nd2 * nvs);
            c2 = fminf(fmaxf(c2, -1.f), 1.f);
            m2p += acosf(fabsf(c2));
        }
    }
    const float m1 = wsum32(m1p);
    const float m2 = wsum32(m2p);
    const bool pick1 = (m1 < m2);
    const float spx = pick1 ? sp1x : sp2x;
    const float spy = pick1 ? sp1y : sp2y;
    const float spz = pick1 ? sp1z : sp2z;

    // ---- near-point mean displacement (radius^2 < 100) ----
    float cntp = 0.f, vxp = 0.f, vyp = 0.f, vzp = 0.f;
#pragma unroll
    for (int j = 0; j < 4; ++j) {
        const float dx = rx[j]-spx, dy = ry[j]-spy, dz = rz[j]-spz;
        const float dd = dx*dx + dy*dy + dz*dz;
        const float nr = ((mj[j] > 0.f) && (dd < 100.0f)) ? 1.0f : 0.0f;
        cntp += nr; vxp += nr*dx; vyp += nr*dy; vzp += nr*dz;
    }
    const float cnt  = wsum32(cntp);
    const float icnt = 1.0f / cnt;   // endpoint itself always near -> cnt >= 1
    const float spdx = wsum32(vxp) * icnt;
    const float spdy = wsum32(vyp) * icnt;
    const float spdz = wsum32(vzp) * icnt;

    if (lane == 0) {
        float* o = out + (size_t)c * 22;
        o[0]  = cx;        o[1]  = cy;        o[2]  = cz;
        o[3]  = a00*iw2;   o[4]  = a01*iw2;   o[5]  = a02*iw2;
        o[6]  = a01*iw2;   o[7]  = a11*iw2;   o[8]  = a12*iw2;
        o[9]  = a02*iw2;   o[10] = a12*iw2;   o[11] = a22*iw2;
        o[12] = v0x;       o[13] = v0y;       o[14] = v0z;
        o[15] = n;
        o[16] = spx;       o[17] = spy;       o[18] = spz;
        o[19] = spdx;      o[20] = spdy;      o[21] = spdz;
    }
}

extern "C" void kernel_launch(void* const* d_in, const int* in_sizes, int n_in,
                              void* d_out, int out_size, void* d_ws, size_t ws_size,
                              hipStream_t stream) {
    (void)in_sizes; (void)n_in; (void)out_size; (void)d_ws; (void)ws_size;
    const float* data    = (const float*)d_in[0];
    const int*   clusts  = (const int*)d_in[1];
    const int*   lengths = (const int*)d_in[2];
    float*       out     = (float*)d_out;
    dim3 grid(N_CLUSTS / 4), block(128);
    clust_geo_kernel<<<grid, block, 0, stream>>>(data, clusts, lengths, out);
}